// HolographicFMLayer_87162066305285
// MI455X (gfx1250) — compile-verified
//
#include <hip/hip_runtime.h>

// CDNA5 / gfx1250: wave32, fp32 matrix path V_WMMA_F32_16X16X4_F32.
// out[b,p,n] = sum_k x[b,i_p,k] * x[b,j_p,(n-k) mod 64]  (circular convolution
// == ifft(fft*fft) of the fp32 reference). Grouped per (batch, j) as GEMMs
// against the circulant of field j. Output-write bound: 145 MB @ 23.3 TB/s.

typedef float v2f __attribute__((ext_vector_type(2)));
typedef float v4f __attribute__((ext_vector_type(4)));
typedef float v8f __attribute__((ext_vector_type(8)));
typedef int   v4i __attribute__((ext_vector_type(4)));

#define NF    24      // fields
#define L     64      // embedding length
#define ROWP  132     // LDS row stride (floats): doubled 128 + 4 pad -> column reads hit distinct banks
#define NPAIR 276     // C(24,2)

#if defined(__has_builtin)
#if __has_builtin(__builtin_amdgcn_global_load_async_to_lds_b128) && \
    __has_builtin(__builtin_amdgcn_s_wait_asynccnt)
#define USE_ASYNC_STAGE 1
#endif
#endif

#ifdef USE_ASYNC_STAGE
// arg0 (probe-confirmed from clang diagnostic): generic 'v4i*'
#define TO_GLB(p) ((v4i*)(p))
// arg1: LDS destination; AS(3) v4i*. Generic LDS VA's low 32 bits == LDS offset.
#define TO_LDS(p) ((__attribute__((address_space(3))) v4i*)(unsigned long long)(unsigned)(unsigned long long)(p))
#endif

__global__ __launch_bounds__(256)
void holo_circconv_wmma(const float* __restrict__ in, float* __restrict__ out)
{
    // X[b] staged doubled along length: XX[f][c] = x_f[c & 63], c in [0,128)
    __shared__ float XX[NF * ROWP];

    const int b   = blockIdx.x;
    const int tid = threadIdx.x;
    const float* xb = in + (size_t)b * (NF * L);

    // ---- stage 24x64 fp32 into LDS (doubled), 384 x b128 transactions ----
    if (tid < 192) {                      // 6 full waves -> uniform EXEC
#pragma unroll
        for (int q = 0; q < 2; ++q) {
            const int i4 = tid + 192 * q;         // float4 index, 16 per row
            const int f  = i4 >> 4;
            const int c  = (i4 & 15) << 2;
            const v4f* g = ((const v4f*)xb) + i4;
#ifdef USE_ASYNC_STAGE
            __builtin_amdgcn_global_load_async_to_lds_b128(
                TO_GLB(g), TO_LDS(&XX[f * ROWP + c]),      0, 0);
            __builtin_amdgcn_global_load_async_to_lds_b128(
                TO_GLB(g), TO_LDS(&XX[f * ROWP + 64 + c]), 0, 0);
#else
            const v4f v = *g;
            *(v4f*)&XX[f * ROWP + c]      = v;
            *(v4f*)&XX[f * ROWP + 64 + c] = v;
#endif
        }
    }
#ifdef USE_ASYNC_STAGE
    __builtin_amdgcn_s_wait_asynccnt(0);
#endif
    __syncthreads();

    const int wave  = tid >> 5;             // 8 wave32s
    const int lane  = tid & 31;
    const int nloc  = lane & 15;            // M (A rows) / N (B,D cols) position
    const int koff  = (lane >> 4) << 1;     // K sub-offset: 0 (lanes 0-15) / 2 (lanes 16-31)
    const int rbase = (lane >> 4) << 3;     // D rows: +8 for upper half-wave

    // 30 (j, m-tile) units per batch: j=1..16 -> 1 tile, j=17..23 -> 2 tiles.
    // Each unit: load A once, run 4 N-tiles x 16 K-steps of WMMA.
    for (int u = wave; u < 30; u += 8) {
        int j, mt;
        if (u < 16) { j = u + 1; mt = 0; }
        else        { const int w = u - 16; j = 17 + (w >> 1); mt = w & 1; }

        // ---- A fragments: rows i = 16*mt + (lane&15), full K=64 depth ----
        int arow = 16 * mt + nloc;
        if (arow > NF - 1) arow = NF - 1;   // clamp padded rows; results discarded
        const float* ap = &XX[arow * ROWP];
        v2f afrag[16];
#pragma unroll
        for (int kb = 0; kb < 16; ++kb)
            afrag[kb] = *(const v2f*)(ap + 4 * kb + koff);   // ds_load_b64

        // ---- per-lane output offsets for the 8 D rows of this unit ----
        const int ibase = 16 * mt + rbase;
        unsigned ooff[8];
#pragma unroll
        for (int r = 0; r < 8; ++r) {
            const int i = ibase + r;
            const int p = 23 * i - (i * (i - 1)) / 2 + (j - i - 1); // valid iff i<j
            ooff[r] = (unsigned)((b * NPAIR + p) * L + nloc);
        }

        // B[k][n] = x_j[(n-k) mod 64] = XX[j][(n+64)-k]  (wrap-free, doubled row)
        const float* bp0 = &XX[j * ROWP + 64 + nloc];
        for (int nt = 0; nt < 4; ++nt) {
            const float* bp = bp0 + 16 * nt;
            v8f acc = {};
#pragma unroll
            for (int kb = 0; kb < 16; ++kb) {
                const int k0 = 4 * kb + koff;
                v2f bf;
                bf.x = bp[-k0];
                bf.y = bp[-k0 - 1];
                // (neg_a, A, neg_b, B, c_mod, C, reuse_a, reuse_b)
                acc = __builtin_amdgcn_wmma_f32_16x16x4_f32(
                    false, afrag[kb], false, bf, (short)0, acc, false, false);
            }
#pragma unroll
            for (int r = 0; r < 8; ++r) {
                const int i = ibase + r;
                if (i < j) out[ooff[r] + 16 * nt] = acc[r];
            }
        }
    }
}

extern "C" void kernel_launch(void* const* d_in, const int* in_sizes, int n_in,
                              void* d_out, int out_size, void* d_ws, size_t ws_size,
                              hipStream_t stream) {
    const float* in = (const float*)d_in[0];
    float* out = (float*)d_out;
    const int batches = in_sizes[0] / (NF * L);   // 2048
    holo_circconv_wmma<<<dim3(batches), dim3(256), 0, stream>>>(in, out);
}